// StochPool_20916490731900
// MI455X (gfx1250) — compile-verified
//
#include <hip/hip_runtime.h>
#include <hip/hip_bf16.h>
#include <math.h>
#include <stdint.h>

typedef __attribute__((ext_vector_type(2))) float v2f;
typedef __attribute__((ext_vector_type(4))) float v4f;
typedef __attribute__((ext_vector_type(8))) float v8f;
typedef __attribute__((ext_vector_type(4))) unsigned int v4u;
typedef __attribute__((ext_vector_type(4))) int v4i;
typedef __attribute__((ext_vector_type(8))) int v8i;

#define B_   64
#define NPG  1024
#define EPG  32768
#define F_   256
#define C_   64
#define N_   (B_ * NPG)     // 65536 nodes
#define E_   (B_ * EPG)     // 2097152 edges
#define K_   (B_ * C_)      // 4096 global clusters

#if __has_builtin(__builtin_amdgcn_tensor_load_to_lds) && \
    __has_builtin(__builtin_amdgcn_s_wait_tensorcnt)
#define USE_TDM 1
#endif

// ---------------------------------------------------------------------------
// Kernel 1: logits = x @ W + b via V_WMMA_F32_16X16X4_F32.
//  - W (256x64 f32, 64KB) is DMA'd into LDS by the Tensor Data Mover
//    (TENSOR_LOAD_TO_LDS, 2D D#), then reshuffled once into a k-pair
//    interleaved layout so every WMMA B operand is a single contiguous
//    ds_load_b64 into an even-aligned VGPR pair (no regrouping moves).
//  - The raw staging region is recycled as the logit spill buffer.
//  - Each wave computes a 16-node x 64-class logit tile (4 N-tiles x 64
//    k-steps of 16x16x4 f32 WMMA); lanes 0..15 then do the per-node
//    softmax max-prob + argmax.
// ---------------------------------------------------------------------------
__global__ __launch_bounds__(256)
void assign_kernel(const float* __restrict__ x, const float* __restrict__ W,
                   const float* __restrict__ bias,
                   int* __restrict__ cOut, float* __restrict__ vOut)
{
    __shared__ float smem[2 * F_ * C_];      // 128 KB
    float* sWraw = smem;                     // TDM destination (row-major W)
    float* sWi   = smem + F_ * C_;           // pair-interleaved W
    float* sLog  = smem;                     // recycled: 8 waves x 16 x 68

    const int lane = threadIdx.x & 31;
    const int wave = threadIdx.x >> 5;
    const int half = lane >> 4;              // 0: K base 0, 1: K base 2
    const int l15  = lane & 15;
    const int nodeBase = blockIdx.x * 128 + wave * 16;

#ifdef USE_TDM
    if (wave == 0) {
        // Build D# (groups 0/1; groups 2/3 + extra group zero for a 2D tensor).
        const unsigned long long ga = (unsigned long long)(uintptr_t)W;
        const unsigned int ldsAddr  = (unsigned int)(uintptr_t)&sWraw[0];
        v4u g0;
        g0[0] = 1u;                                   // count=1 (valid user D#)
        g0[1] = ldsAddr;                              // lds_addr (bytes)
        g0[2] = (unsigned int)ga;                     // global_addr[31:0]
        g0[3] = ((unsigned int)(ga >> 32) & 0x01FFFFFFu) | (2u << 30); // addr[56:32], type=2
        v8i g1;
        g1[0] = (int)(2u << 16);                      // data_size = 4 bytes
        g1[1] = (int)((unsigned)C_ << 16);            // tensor_dim0 = 64
        g1[2] = (int)((unsigned)F_ << 16);            // tensor_dim1 = 256
        g1[3] = (int)((unsigned)C_ << 16);            // tile_dim0 = 64
        g1[4] = (int)F_;                              // tile_dim1 = 256, tile_dim2 = 0
        g1[5] = C_;                                   // tensor_dim0_stride = 64
        g1[6] = 0;
        g1[7] = 0;
        v4i g2 = (v4i)0;
        v4i g3 = (v4i)0;
        v8i g4 = (v8i)0;
        __builtin_amdgcn_tensor_load_to_lds(g0, g1, g2, g3, g4, 0);
        __builtin_amdgcn_s_wait_tensorcnt(0);
    }
    __syncthreads();
    // One-time reshuffle: sWi[(k/2)*128 + c*2 + (k&1)] = W[k][c]
    for (int i = threadIdx.x; i < F_ * C_; i += 256) {
        const int k = i >> 6;
        const int c = i & 63;
        sWi[(k >> 1) * 128 + c * 2 + (k & 1)] = sWraw[i];
    }
#else
    // Fallback: stage directly from global into the interleaved layout.
    for (int i = threadIdx.x; i < F_ * C_; i += 256) {
        const int k = i >> 6;
        const int c = i & 63;
        sWi[(k >> 1) * 128 + c * 2 + (k & 1)] = W[i];
    }
#endif
    __syncthreads();

    const float* xrow = x + (size_t)(nodeBase + l15) * F_;

    v8f acc0 = (v8f)0.0f, acc1 = (v8f)0.0f, acc2 = (v8f)0.0f, acc3 = (v8f)0.0f;

    for (int k0 = 0; k0 < F_; k0 += 4) {
        // A operand: x[node][k0 + 2*half + {0,1}]  (f32 16x16x4 A layout)
        v2f a = *(const v2f*)(xrow + k0 + 2 * half);
        // B operands: contiguous {W[k][c], W[k+1][c]} pairs from interleaved LDS
        const float* wp = sWi + (size_t)((k0 + 2 * half) >> 1) * 128 + l15 * 2;
        {
            v2f b = *(const v2f*)(wp + 0);
            acc0 = __builtin_amdgcn_wmma_f32_16x16x4_f32(false, a, false, b,
                                                         (short)0, acc0, false, false);
        }
        {
            v2f b = *(const v2f*)(wp + 32);
            acc1 = __builtin_amdgcn_wmma_f32_16x16x4_f32(false, a, false, b,
                                                         (short)0, acc1, false, false);
        }
        {
            v2f b = *(const v2f*)(wp + 64);
            acc2 = __builtin_amdgcn_wmma_f32_16x16x4_f32(false, a, false, b,
                                                         (short)0, acc2, false, false);
        }
        {
            v2f b = *(const v2f*)(wp + 96);
            acc3 = __builtin_amdgcn_wmma_f32_16x16x4_f32(false, a, false, b,
                                                         (short)0, acc3, false, false);
        }
    }

    // Spill logits to LDS (recycled raw region; upper half still serves other
    // waves' B reads — regions are disjoint). C/D layout: VGPR r, lane l ->
    // node r + 8*(l>>4), class tile*16 + (l&15).
    float* my = sLog + wave * (16 * 68);
    v8f accs[4] = {acc0, acc1, acc2, acc3};
    #pragma unroll
    for (int t = 0; t < 4; ++t) {
        const int cls = t * 16 + l15;
        const float bv = bias[cls];
        #pragma unroll
        for (int r = 0; r < 8; ++r) {
            const int node = r + 8 * half;
            my[node * 68 + cls] = accs[t][r] + bv;
        }
    }
    __syncthreads();

    // Lanes 0..15: per-node softmax max-prob + argmax (first-max tie break).
    if (lane < 16) {
        const float* row = my + lane * 68;
        float mx = row[0];
        int   am = 0;
        #pragma unroll 8
        for (int i = 1; i < C_; ++i) {
            float t = row[i];
            if (t > mx) { mx = t; am = i; }
        }
        float s = 0.0f;
        #pragma unroll 8
        for (int i = 0; i < C_; ++i) s += expf(row[i] - mx);
        float p = 1.0f / s;                 // softmax value at argmax
        float v = (1.0f - p) + p;           // straight-through forward value
        const int node = nodeBase + lane;
        cOut[node] = am;
        vOut[node] = v;
    }
}

// ---------------------------------------------------------------------------
// Kernel 2: out[gcl] += v[n] * x[n]  (float4 vectorized), vs2[gcl] += v*v
// ---------------------------------------------------------------------------
__global__ __launch_bounds__(256)
void scatter_nodes(const float* __restrict__ x, const int* __restrict__ c,
                   const float* __restrict__ v,
                   float* __restrict__ out, float* __restrict__ vs2)
{
    const int tid = blockIdx.x * 256 + threadIdx.x;   // N_ * 64 threads
    const int n = tid >> 6;
    const int q = tid & 63;                           // float4 chunk id
    if (n >= N_) return;
    const int gcl = (n >> 10) * C_ + c[n];
    const float vn = v[n];
    v4f xv = *(const v4f*)(x + (size_t)n * F_ + q * 4);
    float* o = out + (size_t)gcl * F_ + q * 4;
    atomicAdd(o + 0, vn * xv.x);
    atomicAdd(o + 1, vn * xv.y);
    atomicAdd(o + 2, vn * xv.z);
    atomicAdd(o + 3, vn * xv.w);
    if (q == 0) atomicAdd(vs2 + gcl, vn * vn);
}

// ---------------------------------------------------------------------------
// Kernel 3: per-edge scatter into out_adj; wave32-reduced a2 / ap scalars.
// ---------------------------------------------------------------------------
__global__ __launch_bounds__(256)
void scatter_edges(const int* __restrict__ ei, const float* __restrict__ ew,
                   const int* __restrict__ c, const float* __restrict__ v,
                   float* __restrict__ out_adj, float* __restrict__ a2ap)
{
    const int e = blockIdx.x * 256 + threadIdx.x;
    float ww = 0.0f, app = 0.0f;
    if (e < E_) {
        const int s = ei[e];
        const int d = ei[E_ + e];
        const float wt = ew[e];
        const int ci = c[s];
        const int cj = c[d];
        const int gb = s >> 10;                       // batch = node / NPG
        const float contrib = wt * v[s] * v[d];
        atomicAdd(out_adj + (size_t)gb * (C_ * C_) + ci * C_ + cj, contrib);
        ww = wt * wt;
        if (ci == cj) app = contrib;                  // same graph by construction
    }
    // wave32 tree reduction before the scalar atomics
    #pragma unroll
    for (int off = 16; off > 0; off >>= 1) {
        ww  += __shfl_down(ww,  off, 32);
        app += __shfl_down(app, off, 32);
    }
    if ((threadIdx.x & 31) == 0) {
        atomicAdd(&a2ap[0], ww);
        atomicAdd(&a2ap[1], app);
    }
}

// ---------------------------------------------------------------------------
// Kernel 4: link_loss = sqrt(max(a2 - 2*ap + sum(vs2^2), 0)) / E
// ---------------------------------------------------------------------------
__global__ __launch_bounds__(256)
void finalize(const float* __restrict__ vs2, const float* __restrict__ a2ap,
              float* __restrict__ loss_out, float* __restrict__ zero_out)
{
    __shared__ float red[256];
    float s = 0.0f;
    for (int i = threadIdx.x; i < K_; i += 256) {
        float t = vs2[i];
        s += t * t;
    }
    red[threadIdx.x] = s;
    __syncthreads();
    #pragma unroll
    for (int st = 128; st > 0; st >>= 1) {
        if (threadIdx.x < st) red[threadIdx.x] += red[threadIdx.x + st];
        __syncthreads();
    }
    if (threadIdx.x == 0) {
        float val = a2ap[0] - 2.0f * a2ap[1] + red[0];
        loss_out[0] = sqrtf(fmaxf(val, 0.0f)) / (float)E_;
        zero_out[0] = 0.0f;
    }
}

// ---------------------------------------------------------------------------
// Kernel 5: deterministic metadata (indices written as float output dtype)
// ---------------------------------------------------------------------------
__global__ __launch_bounds__(256)
void fill_meta(float* __restrict__ ei_out, float* __restrict__ batch_out,
               float* __restrict__ bptr_out)
{
    const int idx = blockIdx.x * 256 + threadIdx.x;
    if (idx < B_ * C_ * C_) {
        const int bb  = idx >> 12;       // / (C_*C_)
        const int rem = idx & 4095;
        ei_out[idx]                = (float)(bb * C_ + (rem >> 6));
        ei_out[B_ * C_ * C_ + idx] = (float)(bb * C_ + (rem & 63));
    }
    if (idx < K_)     batch_out[idx] = (float)(idx >> 6);
    if (idx < B_ + 1) bptr_out[idx]  = (float)(idx * C_);
}

// ---------------------------------------------------------------------------
extern "C" void kernel_launch(void* const* d_in, const int* in_sizes, int n_in,
                              void* d_out, int out_size, void* d_ws, size_t ws_size,
                              hipStream_t stream)
{
    (void)in_sizes; (void)n_in; (void)out_size; (void)ws_size;

    const float* x    = (const float*)d_in[0];   // [N, F]
    const int*   ei   = (const int*)  d_in[1];   // [2, E]
    const float* ew   = (const float*)d_in[2];   // [E]
    //            batch d_in[3], batch_ptr d_in[4] are derivable (NPG fixed)
    const float* W    = (const float*)d_in[5];   // [F, C]
    const float* bias = (const float*)d_in[6];   // [C]

    float* out      = (float*)d_out;                       // [K_, F_]
    float* out_ei   = out + (size_t)K_ * F_;               // [2, B*C*C]
    float* out_adj  = out_ei + 2 * B_ * C_ * C_;           // [B*C*C]
    float* loss     = out_adj + B_ * C_ * C_;              // [1]
    float* zero     = loss + 1;                            // [1]
    float* batch_o  = zero + 1;                            // [K_]
    float* bptr_o   = batch_o + K_;                        // [B_+1]

    int*   cbuf = (int*)d_ws;                              // [N_]
    float* vbuf = (float*)(cbuf + N_);                     // [N_]
    float* vs2  = vbuf + N_;                               // [K_]
    float* a2ap = vs2 + K_;                                // [2]

    // zero the accumulators every call (graph-capture-safe memsets)
    (void)hipMemsetAsync(out,     0, (size_t)K_ * F_ * sizeof(float), stream);
    (void)hipMemsetAsync(out_adj, 0, (size_t)B_ * C_ * C_ * sizeof(float), stream);
    (void)hipMemsetAsync(vs2,     0, (K_ + 2) * sizeof(float), stream);

    assign_kernel<<<N_ / 128, 256, 0, stream>>>(x, W, bias, cbuf, vbuf);
    scatter_nodes<<<(N_ * 64) / 256, 256, 0, stream>>>(x, cbuf, vbuf, out, vs2);
    scatter_edges<<<(E_ + 255) / 256, 256, 0, stream>>>(ei, ew, cbuf, vbuf, out_adj, a2ap);
    finalize<<<1, 256, 0, stream>>>(vs2, a2ap, loss, zero);
    fill_meta<<<(B_ * C_ * C_ + 255) / 256, 256, 0, stream>>>(out_ei, batch_o, bptr_o);
}